// SelfModifyingTitans_56753697849855
// MI455X (gfx1250) — compile-verified
//
#include <hip/hip_runtime.h>
#include <math.h>

typedef __attribute__((ext_vector_type(2))) float v2f;
typedef __attribute__((ext_vector_type(8))) float v8f;

#define AS1 __attribute__((address_space(1)))
typedef const AS1 float* gcf_t;   // global const float*
typedef AS1 float*       gf_t;    // global float*
typedef const AS1 v2f*   gcv2_t;  // global const float2*

__device__ __forceinline__ gcf_t as_gc(const float* p) { return (gcf_t)p; }
__device__ __forceinline__ gf_t  as_g(float* p)        { return (gf_t)p; }

namespace {
constexpr int DD  = 256;          // model dim
constexpr int CH  = 64;           // chunk length
constexpr int TT  = 4096;         // sequence length
constexpr int NCK = TT / CH;      // 64 chunks
constexpr int W2  = DD * DD;      // 65536 floats per weight matrix
constexpr int ACT = CH * DD;      // 16384 floats per activation tile
constexpr int RPAD = 258;         // LDS row stride for row-major staged tile
constexpr int TPAD = 66;          // LDS row stride for transposed staged tile
constexpr int SBUF = 16896;       // 256*66 floats (also >= 64*258)
constexpr size_t PER_BATCH = 10 * (size_t)W2 + 1024 + 8 * (size_t)ACT + 256;
}

__device__ __forceinline__ float geluf(float z) {
    return 0.5f * z * (1.0f + erff(z * 0.70710678118654752f));
}
__device__ __forceinline__ float dgeluf(float z) {
    float cdf = 0.5f * (1.0f + erff(z * 0.70710678118654752f));
    float pdf = expf(-0.5f * z * z) * 0.39894228040143267f;
    return cdf + z * pdf;
}
__device__ __forceinline__ v8f wmma4(v2f a, v2f b, v8f c) {
    return __builtin_amdgcn_wmma_f32_16x16x4_f32(false, a, false, b, (short)0, c,
                                                 false, false);
}

// ---------------------------------------------------------------------------
// LDS staging helpers (leading sync protects buffer reuse, trailing sync
// publishes the staged tile).
// ---------------------------------------------------------------------------
// row-major stage: s[t*258 + k] = g[t*256 + k]
__device__ void stageA(const float* __restrict__ gp, float* __restrict__ s) {
    gcf_t g = as_gc(gp);
    __syncthreads();
    for (int i = threadIdx.x; i < ACT / 2; i += 256) {
        const int t  = i >> 7;
        const int k2 = (i & 127) << 1;
        *(v2f*)(s + t * RPAD + k2) = *(gcv2_t)(g + (i << 1));
    }
    __syncthreads();
}
// transposed stage with eta: s[m*66 + t] = g[t*256 + m] * eta[t]
__device__ void stageT_eta(const float* __restrict__ gp, const float* __restrict__ eta,
                           float* __restrict__ s) {
    gcf_t g = as_gc(gp);
    __syncthreads();
    for (int i = threadIdx.x; i < ACT; i += 256) {
        const int t = i >> 8, m = i & 255;
        s[m * TPAD + t] = g[i] * eta[t];
    }
    __syncthreads();
}
// transposed stage: s[m*66 + t] = g[t*256 + m]
__device__ void stageT(const float* __restrict__ gp, float* __restrict__ s) {
    gcf_t g = as_gc(gp);
    __syncthreads();
    for (int i = threadIdx.x; i < ACT; i += 256) {
        const int t = i >> 8, m = i & 255;
        s[m * TPAD + t] = g[i];
    }
    __syncthreads();
}

// ---------------------------------------------------------------------------
// WMMA micro-GEMMs (f32 16x16x4). 8 waves, wave32.
// A fragment (16x4): reg r holds K = k0 + 2*(lane>>4) + r, M = m0 + (lane&15)
// B fragment (4x16): reg r holds K = k0 + 2*(lane>>4) + r, N = n0 + (lane&15)
// C fragment: reg r -> row m0 + r + 8*(lane>>4), col n0 + (lane&15)
// ---------------------------------------------------------------------------

// C[64][256] = A_lds @ W^T : wave owns 2 N-blocks x 4 M-block accumulators;
// each W row is fetched exactly once, A fragments come from LDS.
__device__ void gemm64_AWt_core(const float* __restrict__ sA, const float* __restrict__ Wp,
                                float* __restrict__ Cp) {
    gcf_t W = as_gc(Wp);
    gf_t  C = as_g(Cp);
    const int lane = threadIdx.x & 31;
    const int wave = threadIdx.x >> 5;
    const int half = lane >> 4;
    const int l16  = lane & 15;
    for (int j = 0; j < 2; ++j) {
        const int n0 = ((wave << 1) | j) << 4;
        v8f acc0 = {}, acc1 = {}, acc2 = {}, acc3 = {};
        gcf_t brow = W + (n0 + l16) * DD + 2 * half;
        const float* a0 = sA + l16 * RPAD + 2 * half;
        for (int k0 = 0; k0 < DD; k0 += 4) {
            v2f b  = *(gcv2_t)(brow + k0);
            v2f A0 = *(const v2f*)(a0 + k0);
            v2f A1 = *(const v2f*)(a0 + 16 * RPAD + k0);
            v2f A2 = *(const v2f*)(a0 + 32 * RPAD + k0);
            v2f A3 = *(const v2f*)(a0 + 48 * RPAD + k0);
            acc0 = wmma4(A0, b, acc0);
            acc1 = wmma4(A1, b, acc1);
            acc2 = wmma4(A2, b, acc2);
            acc3 = wmma4(A3, b, acc3);
        }
        gf_t ccol    = C + n0 + l16;
        const int rb = 8 * half;
#pragma unroll
        for (int r = 0; r < 8; ++r) {
            ccol[(r + rb) * DD]        = acc0[r];
            ccol[(16 + r + rb) * DD]   = acc1[r];
            ccol[(32 + r + rb) * DD]   = acc2[r];
            ccol[(48 + r + rb) * DD]   = acc3[r];
        }
    }
}

// C[64][256] = G_lds @ W   (non-transposed weight access)
__device__ void gemm64_AW_core(const float* __restrict__ sG, const float* __restrict__ Wp,
                               float* __restrict__ Cp) {
    gcf_t W = as_gc(Wp);
    gf_t  C = as_g(Cp);
    const int lane = threadIdx.x & 31;
    const int wave = threadIdx.x >> 5;
    const int half = lane >> 4;
    const int l16  = lane & 15;
    for (int j = 0; j < 2; ++j) {
        const int n0 = ((wave << 1) | j) << 4;
        v8f acc0 = {}, acc1 = {}, acc2 = {}, acc3 = {};
        gcf_t bcol = W + n0 + l16;
        const float* a0 = sG + l16 * RPAD + 2 * half;
        for (int k0 = 0; k0 < DD; k0 += 4) {
            const int kk = k0 + 2 * half;
            v2f b;
            b.x = bcol[kk * DD];
            b.y = bcol[(kk + 1) * DD];
            v2f A0 = *(const v2f*)(a0 + k0);
            v2f A1 = *(const v2f*)(a0 + 16 * RPAD + k0);
            v2f A2 = *(const v2f*)(a0 + 32 * RPAD + k0);
            v2f A3 = *(const v2f*)(a0 + 48 * RPAD + k0);
            acc0 = wmma4(A0, b, acc0);
            acc1 = wmma4(A1, b, acc1);
            acc2 = wmma4(A2, b, acc2);
            acc3 = wmma4(A3, b, acc3);
        }
        gf_t ccol    = C + n0 + l16;
        const int rb = 8 * half;
#pragma unroll
        for (int r = 0; r < 8; ++r) {
            ccol[(r + rb) * DD]        = acc0[r];
            ccol[(16 + r + rb) * DD]   = acc1[r];
            ccol[(32 + r + rb) * DD]   = acc2[r];
            ccol[(48 + r + rb) * DD]   = acc3[r];
        }
    }
}

// W[256][256] = abar*W - A_s^T @ B  with both operands staged transposed in LDS
// (eta already folded into A_s). Wave owns 2 M-blocks; 16 A fragments per
// M-block preloaded into registers and reused across all 16 N-tiles.
__device__ void gemmTT_core(float* __restrict__ Wp, const float* __restrict__ sAt,
                            const float* __restrict__ sBt, float abar) {
    gf_t W = as_g(Wp);
    const int lane = threadIdx.x & 31;
    const int wave = threadIdx.x >> 5;
    const int half = lane >> 4;
    const int l16  = lane & 15;
#pragma unroll
    for (int mi = 0; mi < 2; ++mi) {
        const int m0 = (wave + mi * 8) << 4;
        v2f af[16];
        const float* ap = sAt + (m0 + l16) * TPAD + 2 * half;
#pragma unroll
        for (int kk = 0; kk < 16; ++kk) af[kk] = *(const v2f*)(ap + kk * 4);
        for (int nb = 0; nb < 16; ++nb) {
            const int n0 = nb << 4;
            v8f acc = {};
            const float* bp = sBt + (n0 + l16) * TPAD + 2 * half;
#pragma unroll
            for (int kk = 0; kk < 16; ++kk) {
                v2f b = *(const v2f*)(bp + kk * 4);
                acc = wmma4(af[kk], b, acc);
            }
            const int mbase = m0 + 8 * half;
#pragma unroll
            for (int r = 0; r < 8; ++r) {
                const int idx = (mbase + r) * DD + n0 + l16;
                W[idx] = abar * W[idx] - acc[r];
            }
        }
    }
}

// wrappers (staging + core); caller syncs after before consuming C
__device__ void gemm64_AWt(const float* A, const float* W, float* C, float* sS) {
    stageA(A, sS);
    gemm64_AWt_core(sS, W, C);
}
__device__ void gemm64_AW(const float* G, const float* W, float* C, float* sS) {
    stageA(G, sS);
    gemm64_AW_core(sS, W, C);
}
__device__ void gemmTT_update(float* W, const float* A, const float* B,
                              const float* eta, float abar, float* sXp, float* sYp) {
    stageT_eta(A, eta, sXp);
    stageT(B, sYp);
    gemmTT_core(W, sXp, sYp, abar);
    __syncthreads();
}

// ---------------------------------------------------------------------------
// elementwise / reduction helpers
// ---------------------------------------------------------------------------
__device__ void ew_gelu(const float* __restrict__ Zp, float* __restrict__ Hp) {
    gcf_t Z = as_gc(Zp);
    gf_t  H = as_g(Hp);
    for (int i = threadIdx.x; i < ACT; i += blockDim.x) H[i] = geluf(Z[i]);
    __syncthreads();
}

__device__ void l2norm_rows(float* __restrict__ Ap) {
    gf_t A = as_g(Ap);
    if (threadIdx.x < CH) {
        gf_t r = A + threadIdx.x * DD;
        float s = 0.f;
        for (int i = 0; i < DD; ++i) s += r[i] * r[i];
        float n = fmaxf(sqrtf(s), 1e-6f);
        float inv = 1.0f / n;
        for (int i = 0; i < DD; ++i) r[i] *= inv;
    }
    __syncthreads();
}

// OUT = X + gelu(X @ w2^T) @ w1^T
__device__ void memfwd(const float* __restrict__ X, const float* __restrict__ w1,
                       const float* __restrict__ w2, float* __restrict__ Z,
                       float* __restrict__ H, float* __restrict__ OUT, float* sS) {
    gemm64_AWt(X, w2, Z, sS);
    __syncthreads();
    ew_gelu(Z, H);
    gemm64_AWt(H, w1, OUT, sS);
    __syncthreads();
    {
        gcf_t Xg = as_gc(X);
        gf_t  Og = as_g(OUT);
        for (int i = threadIdx.x; i < ACT; i += blockDim.x) Og[i] += Xg[i];
    }
    __syncthreads();
}

// raw[t] = dot(X[t], wskip) + dot(gelu(X @ w2^T)[t], w1)     (1-dim head)
__device__ void smallraw(const float* __restrict__ X, const float* __restrict__ w1,
                         const float* __restrict__ wskip, const float* __restrict__ w2,
                         float* __restrict__ Z, float* __restrict__ H,
                         float* __restrict__ raw, float* sS) {
    gemm64_AWt(X, w2, Z, sS);
    __syncthreads();
    ew_gelu(Z, H);
    if (threadIdx.x < CH) {
        gcf_t xr = as_gc(X) + threadIdx.x * DD;
        gcf_t hr = as_gc(H) + threadIdx.x * DD;
        gcf_t ws1 = as_gc(w1);
        gcf_t wsk = as_gc(wskip);
        float f = 0.f;
        for (int i = 0; i < DD; ++i) f += xr[i] * wsk[i] + hr[i] * ws1[i];
        raw[threadIdx.x] = f;
    }
    __syncthreads();
}

// big-state update: w1 <- abar*w1 - gW1 ; w2 <- abar*w2 - gW2
__device__ void update_big(float* __restrict__ w1, float* __restrict__ w2,
                           const float* __restrict__ K, const float* __restrict__ V,
                           const float* __restrict__ eta, float abar,
                           float* __restrict__ Z, float* __restrict__ H,
                           float* __restrict__ F, float* __restrict__ G,
                           float* __restrict__ P, float* sX, float* sY) {
    gemm64_AWt(K, w2, Z, sX);      // z = k @ w2^T
    __syncthreads();
    ew_gelu(Z, H);                 // h = gelu(z)
    gemm64_AWt(H, w1, F, sX);      // f = h @ w1^T
    __syncthreads();
    {
        gcf_t Kg = as_gc(K);
        gcf_t Fg = as_gc(F);
        gcf_t Vg = as_gc(V);
        gf_t  Gg = as_g(G);
        for (int i = threadIdx.x; i < ACT; i += blockDim.x)
            Gg[i] = 2.0f * (Kg[i] + Fg[i] - Vg[i]);   // g = 2*(k + f - v)
    }
    __syncthreads();
    gemm64_AW(G, w1, P, sX);       // pre = g @ w1  (reads OLD w1)
    __syncthreads();
    {
        gcf_t Zg = as_gc(Z);
        gf_t  Pg = as_g(P);
        for (int i = threadIdx.x; i < ACT; i += blockDim.x) Pg[i] *= dgeluf(Zg[i]);
    }
    __syncthreads();
    gemmTT_update(w1, G, H, eta, abar, sX, sY);   // w1 update
    gemmTT_update(w2, P, K, eta, abar, sX, sY);   // w2 update
}

// eta/alpha-state update (1-dim head with skip)
__device__ void update_small(float* __restrict__ w1, float* __restrict__ w2,
                             float* __restrict__ wskip, const float* __restrict__ K,
                             const float* __restrict__ rowsumV,
                             const float* __restrict__ eta, float abar,
                             float* __restrict__ Z, float* __restrict__ H,
                             float* __restrict__ P, float* __restrict__ gs,
                             float* sX, float* sY) {
    gemm64_AWt(K, w2, Z, sX);
    __syncthreads();
    ew_gelu(Z, H);
    if (threadIdx.x < CH) {                    // g[t] = 2*(D*f[t] - sum_o v[t][o])
        const int t = threadIdx.x;
        gcf_t kr = as_gc(K) + t * DD;
        gcf_t hr = as_gc(H) + t * DD;
        gcf_t ws1 = as_gc(w1);
        gcf_t wsk = as_gc(wskip);
        float f = 0.f;
        for (int i = 0; i < DD; ++i) f += kr[i] * wsk[i] + hr[i] * ws1[i];
        gs[t] = 2.0f * ((float)DD * f - rowsumV[t]);
    }
    __syncthreads();
    {
        gcf_t Zg = as_gc(Z);
        gcf_t w1g = as_gc(w1);
        gf_t  Pg = as_g(P);
        for (int i = threadIdx.x; i < ACT; i += blockDim.x) {   // pre = g (x) w1 * dgelu(z)
            const int t = i >> 8, h = i & 255;
            Pg[i] = gs[t] * w1g[h] * dgeluf(Zg[i]);
        }
    }
    __syncthreads();
    {   // per-column reductions for w1 and wskip updates
        const int h = threadIdx.x;             // 256 threads == 256 columns
        gcf_t Hg = as_gc(H);
        gcf_t Kg = as_gc(K);
        float s1 = 0.f, s2 = 0.f;
        for (int t = 0; t < CH; ++t) {
            const float ge = gs[t] * eta[t];
            s1 += ge * Hg[t * DD + h];
            s2 += ge * Kg[t * DD + h];
        }
        const float nw1 = abar * w1[h] - s1;
        const float nsk = abar * wskip[h] - s2;
        w1[h] = nw1;
        wskip[h] = nsk;
    }
    __syncthreads();
    gemmTT_update(w2, P, K, eta, abar, sX, sY);
}

__device__ void copyN(float* __restrict__ dstp, const float* __restrict__ srcp, int n) {
    gf_t  dst = as_g(dstp);
    gcf_t src = as_gc(srcp);
    for (int i = threadIdx.x; i < n; i += blockDim.x) dst[i] = src[i];
}

// ---------------------------------------------------------------------------
// persistent kernel: one workgroup (8 waves) per batch element
// ---------------------------------------------------------------------------
__global__ __launch_bounds__(256) void titans_kernel(
    const float* __restrict__ x,
    const float* __restrict__ i_kw1, const float* __restrict__ i_kw2,
    const float* __restrict__ i_vw1, const float* __restrict__ i_vw2,
    const float* __restrict__ i_qw1, const float* __restrict__ i_qw2,
    const float* __restrict__ i_ew1, const float* __restrict__ i_ew2,
    const float* __restrict__ i_eskip,
    const float* __restrict__ i_aw1, const float* __restrict__ i_aw2,
    const float* __restrict__ i_askip,
    const float* __restrict__ i_mw1, const float* __restrict__ i_mw2,
    float* __restrict__ out, float* __restrict__ ws) {
    __shared__ float sX[SBUF];     // staged A operand (row-major or transposed)
    __shared__ float sY[SBUF];     // staged B operand (transposed)

    const int b = blockIdx.x;
    float* wsb = ws + (size_t)b * PER_BATCH;

    float* kw1 = wsb + 0 * W2;  float* kw2 = wsb + 1 * W2;
    float* vw1 = wsb + 2 * W2;  float* vw2 = wsb + 3 * W2;
    float* qw1 = wsb + 4 * W2;  float* qw2 = wsb + 5 * W2;
    float* mw1 = wsb + 6 * W2;  float* mw2 = wsb + 7 * W2;
    float* ew2 = wsb + 8 * W2;  float* aw2 = wsb + 9 * W2;
    float* smallw = wsb + 10 * W2;
    float* ew1 = smallw;        float* eskip = smallw + 256;
    float* aw1 = smallw + 512;  float* askip = smallw + 768;
    float* act = smallw + 1024;
    float* K = act + 0 * ACT;  float* V = act + 1 * ACT;
    float* Q = act + 2 * ACT;  float* H = act + 3 * ACT;
    float* Z = act + 4 * ACT;  float* G = act + 5 * ACT;
    float* P = act + 6 * ACT;  float* F = act + 7 * ACT;
    float* sv  = act + 8 * ACT;   // [0,64) eta, [64,128) scratch g, [128,192) rowsumV, [192] abar
    float* eta = sv;
    float* gs  = sv + 64;
    float* rsv = sv + 128;

    // initialize the six mutable states from the const inputs
    copyN(kw1, i_kw1, W2);  copyN(kw2, i_kw2, W2);
    copyN(vw1, i_vw1, W2);  copyN(vw2, i_vw2, W2);
    copyN(qw1, i_qw1, W2);  copyN(qw2, i_qw2, W2);
    copyN(mw1, i_mw1, W2);  copyN(mw2, i_mw2, W2);
    copyN(ew2, i_ew2, W2);  copyN(aw2, i_aw2, W2);
    copyN(ew1, i_ew1, DD);  copyN(eskip, i_eskip, DD);
    copyN(aw1, i_aw1, DD);  copyN(askip, i_askip, DD);
    __syncthreads();

    for (int c = 0; c < NCK; ++c) {
        const float* X = x + ((size_t)b * TT + (size_t)c * CH) * DD;

        // ---- forward ----
        memfwd(X, kw1, kw2, Z, H, K, sX);   l2norm_rows(K);      // k
        memfwd(X, vw1, vw2, Z, H, V, sX);                        // v
        if (threadIdx.x < CH) {                                  // rowsum(v) for small heads
            gcf_t vr = as_gc(V) + threadIdx.x * DD;
            float s = 0.f;
            for (int i = 0; i < DD; ++i) s += vr[i];
            rsv[threadIdx.x] = s;
        }
        __syncthreads();
        memfwd(X, qw1, qw2, Z, H, Q, sX);   l2norm_rows(Q);      // q

        smallraw(X, ew1, eskip, ew2, Z, H, gs, sX);              // eta head
        if (threadIdx.x < CH) {
            const float r = gs[threadIdx.x];
            const float sp = (r > 20.f) ? r : log1pf(expf(r));
            eta[threadIdx.x] = sp * 0.001f;
        }
        __syncthreads();

        smallraw(X, aw1, askip, aw2, Z, H, gs, sX);              // alpha head
        if (threadIdx.x < CH) gs[threadIdx.x] = 1.0f / (1.0f + expf(-gs[threadIdx.x]));
        __syncthreads();
        if (threadIdx.x == 0) {
            float p = 1.0f;
            for (int t = 0; t < CH; ++t) p *= gs[t];
            sv[192] = p;                                          // abar
        }
        __syncthreads();
        const float abar = sv[192];

        memfwd(Q, mw1, mw2, Z, H, F, sX);                        // o = mem(q)
        {
            gf_t  op = as_g(out) + ((size_t)b * TT + (size_t)c * CH) * DD;
            gcf_t Fg = as_gc(F);
            for (int i = threadIdx.x; i < ACT; i += blockDim.x) op[i] = Fg[i];
        }
        __syncthreads();

        // ---- self-modification (all use old per-state weights + shared k,v,eta,abar) ----
        update_big(kw1, kw2, K, V, eta, abar, Z, H, F, G, P, sX, sY);
        update_big(vw1, vw2, K, V, eta, abar, Z, H, F, G, P, sX, sY);
        update_big(qw1, qw2, K, V, eta, abar, Z, H, F, G, P, sX, sY);
        update_small(ew1, ew2, eskip, K, rsv, eta, abar, Z, H, P, gs, sX, sY);
        update_small(aw1, aw2, askip, K, rsv, eta, abar, Z, H, P, gs, sX, sY);
        update_big(mw1, mw2, K, V, eta, abar, Z, H, F, G, P, sX, sY);
    }
}

extern "C" void kernel_launch(void* const* d_in, const int* in_sizes, int n_in,
                              void* d_out, int out_size, void* d_ws, size_t ws_size,
                              hipStream_t stream) {
    (void)in_sizes; (void)n_in; (void)out_size; (void)ws_size;
    const float* x      = (const float*)d_in[0];
    const float* kw1    = (const float*)d_in[1];
    const float* kw2    = (const float*)d_in[2];
    const float* vw1    = (const float*)d_in[3];
    const float* vw2    = (const float*)d_in[4];
    const float* qw1    = (const float*)d_in[5];
    const float* qw2    = (const float*)d_in[6];
    const float* ew1    = (const float*)d_in[7];
    const float* ew2    = (const float*)d_in[8];
    const float* eskip  = (const float*)d_in[9];
    const float* aw1    = (const float*)d_in[10];
    const float* aw2    = (const float*)d_in[11];
    const float* askip  = (const float*)d_in[12];
    const float* mw1    = (const float*)d_in[13];
    const float* mw2    = (const float*)d_in[14];

    titans_kernel<<<dim3(8), dim3(256), 0, stream>>>(
        x, kw1, kw2, vw1, vw2, qw1, qw2, ew1, ew2, eskip,
        aw1, aw2, askip, mw1, mw2, (float*)d_out, (float*)d_ws);
}